// CantorAttention_35837207118398
// MI455X (gfx1250) — compile-verified
//
#include <hip/hip_runtime.h>
#include <math.h>

typedef __attribute__((ext_vector_type(2))) float v2f;
typedef __attribute__((ext_vector_type(8))) float v8f;

// Problem constants (match reference)
constexpr int BB  = 2;
constexpr int SS  = 2048;
constexpr int DIM = 1024;
constexpr int NH  = 16;
constexpr int HD  = 64;
constexpr int KNB = 64;                 // neighbors per query
constexpr float SCALE = 0.125f;         // 1/sqrt(64)

// ---------------------------------------------------------------------------
// GEMM: C[M,N] = A[M,Kd] @ W[N,Kd]^T + bias[N]   (torch Linear semantics)
// fp32 WMMA 16x16x4. Block tile 128x128, KC=32, 256 threads (8 waves).
// Wave w computes rows [w*16, w*16+16) x all 128 cols of the block tile.
// LDS tiles stored [row][KC+4] so each lane's WMMA fragment is one b64 read.
// ---------------------------------------------------------------------------
constexpr int BM = 128, BN = 128, KC = 32, LP = KC + 4;   // LDS pitch (16B aligned rows)

__global__ __launch_bounds__(256)
void gemm_wmma_f32(const float* __restrict__ A, const float* __restrict__ W,
                   const float* __restrict__ bias, float* __restrict__ C,
                   int M, int N, int Kd)
{
    __shared__ float As[BM * LP];
    __shared__ float Bs[BN * LP];

    const int tid  = threadIdx.x;
    const int lane = tid & 31;
    const int wv   = tid >> 5;
    const int mBase = blockIdx.y * BM;
    const int nBase = blockIdx.x * BN;

    v8f acc[8];
#pragma unroll
    for (int j = 0; j < 8; ++j) acc[j] = (v8f)0.0f;

    // per-lane fragment addressing (16x4 A tile / 4x16 B tile):
    // lanes 0-15 take k+0,k+1 ; lanes 16-31 take k+2,k+3 (row = lane&15)
    const int fr = lane & 15;
    const int fk = (lane >> 4) << 1;

    for (int kc = 0; kc < Kd; kc += KC) {
        // cooperative load: 128x32 floats each for A and W tiles (float4 moves)
#pragma unroll
        for (int i = 0; i < 4; ++i) {
            int f   = tid + i * 256;          // 0..1023 float4 slots
            int row = f >> 3;
            int c4  = (f & 7) << 2;
            *(float4*)&As[row * LP + c4] =
                *(const float4*)&A[(size_t)(mBase + row) * Kd + kc + c4];
            *(float4*)&Bs[row * LP + c4] =
                *(const float4*)&W[(size_t)(nBase + row) * Kd + kc + c4];
        }
        __syncthreads();

#pragma unroll
        for (int kk = 0; kk < KC; kk += 4) {
            v2f a = *(const v2f*)&As[(wv * 16 + fr) * LP + kk + fk];
#pragma unroll
            for (int j = 0; j < 8; ++j) {
                v2f b = *(const v2f*)&Bs[(j * 16 + fr) * LP + kk + fk];
                acc[j] = __builtin_amdgcn_wmma_f32_16x16x4_f32(
                    false, a, false, b, (short)0, acc[j], false, false);
            }
        }
        __syncthreads();
    }

    // C layout: VGPR i -> row i (lanes 0-15) / row i+8 (lanes 16-31), col = lane&15
    const int rbase = mBase + wv * 16 + ((lane >> 4) << 3);
#pragma unroll
    for (int j = 0; j < 8; ++j) {
        int col  = nBase + j * 16 + (lane & 15);
        float bb = bias[col];
#pragma unroll
        for (int i = 0; i < 8; ++i) {
            C[(size_t)(rbase + i) * N + col] = acc[j][i] + bb;
        }
    }
}

// ---------------------------------------------------------------------------
// Cantor-routed attention: one wave32 per (b, h, q).
// qkv layout: [B, S, 3*DIM] ; q at +h*64, k at +DIM+h*64, v at +2*DIM+h*64.
// Score pass: lane owns neighbors j=lane and j=lane+32, streams K rows (f4).
// Softmax: wave32 butterfly max/sum.  V pass: coalesced f2 gathers, weights
// broadcast from LDS.  Output written as [B, S, DIM] (head-major inside row).
// ---------------------------------------------------------------------------
__global__ __launch_bounds__(256)
void attn_cantor(const float* __restrict__ qkv, const int* __restrict__ routes,
                 float* __restrict__ out)
{
    __shared__ float qs[8][HD];
    __shared__ float wsm[8][KNB];
    __shared__ int   rsm[8][KNB];

    const int lane = threadIdx.x & 31;
    const int w    = threadIdx.x >> 5;
    const int g    = blockIdx.x * 8 + w;          // global query id over B*H*S
    const int b    = g / (NH * SS);
    const int rem  = g - b * (NH * SS);
    const int h    = rem / SS;
    const int s    = rem - h * SS;

    const size_t rowStride = (size_t)3 * DIM;
    const size_t qbase = (size_t)(b * SS + s) * rowStride + h * HD;

    // stage q row + routes into LDS
    *(float2*)&qs[w][2 * lane] = *(const float2*)&qkv[qbase + 2 * lane];
    const int r0 = routes[s * KNB + lane];
    const int r1 = routes[s * KNB + 32 + lane];
    rsm[w][lane]      = r0;
    rsm[w][lane + 32] = r1;
    __syncthreads();

    // scores: dot(q, k[r0]) and dot(q, k[r1])
    const size_t kb0 = (size_t)(b * SS + r0) * rowStride + DIM + h * HD;
    const size_t kb1 = (size_t)(b * SS + r1) * rowStride + DIM + h * HD;
    float a0 = 0.0f, a1 = 0.0f;
#pragma unroll
    for (int d = 0; d < HD; d += 4) {
        float4 qd = *(const float4*)&qs[w][d];
        float4 ka = *(const float4*)&qkv[kb0 + d];
        float4 kb = *(const float4*)&qkv[kb1 + d];
        a0 += qd.x * ka.x + qd.y * ka.y + qd.z * ka.z + qd.w * ka.w;
        a1 += qd.x * kb.x + qd.y * kb.y + qd.z * kb.z + qd.w * kb.w;
    }
    float s0 = a0 * SCALE;
    float s1 = a1 * SCALE;

    // softmax over the 64 scores held 2-per-lane across the wave
    float m = fmaxf(s0, s1);
#pragma unroll
    for (int off = 16; off > 0; off >>= 1)
        m = fmaxf(m, __shfl_xor(m, off, 32));
    float e0 = __expf(s0 - m);
    float e1 = __expf(s1 - m);
    float sum = e0 + e1;
#pragma unroll
    for (int off = 16; off > 0; off >>= 1)
        sum += __shfl_xor(sum, off, 32);
    float inv = 1.0f / sum;
    wsm[w][lane]      = e0 * inv;
    wsm[w][lane + 32] = e1 * inv;
    __syncthreads();

    // weighted sum of V rows; lane owns output dims 2*lane, 2*lane+1
    float ox = 0.0f, oy = 0.0f;
    const size_t vcol = (size_t)2 * DIM + h * HD + 2 * lane;
#pragma unroll 4
    for (int j = 0; j < KNB; ++j) {
        int   r  = rsm[w][j];
        float wt = wsm[w][j];
        float2 vv = *(const float2*)&qkv[(size_t)(b * SS + r) * rowStride + vcol];
        ox += wt * vv.x;
        oy += wt * vv.y;
    }

    float2 o2 = make_float2(ox, oy);
    *(float2*)&out[(size_t)(b * SS + s) * DIM + h * HD + 2 * lane] = o2;
}

// ---------------------------------------------------------------------------
// Launch: qkv GEMM -> attention -> output GEMM
// ws layout: [0, 48MB) qkv fp32 [B*S, 3*DIM] ; [48MB, 64MB) attn out [B*S, DIM]
// ---------------------------------------------------------------------------
extern "C" void kernel_launch(void* const* d_in, const int* in_sizes, int n_in,
                              void* d_out, int out_size, void* d_ws, size_t ws_size,
                              hipStream_t stream)
{
    (void)in_sizes; (void)n_in; (void)out_size; (void)ws_size;

    const float* x     = (const float*)d_in[0];
    const float* w_qkv = (const float*)d_in[1];
    const float* b_qkv = (const float*)d_in[2];
    const float* w_out = (const float*)d_in[3];
    const float* b_out = (const float*)d_in[4];
    const int*   routes = (const int*)d_in[5];

    float* qkv  = (float*)d_ws;
    float* attn = (float*)((char*)d_ws + (size_t)BB * SS * 3 * DIM * sizeof(float));

    const int M = BB * SS;           // 4096

    // QKV projection: [4096,1024] @ [1024,3072] + b
    gemm_wmma_f32<<<dim3((3 * DIM) / BN, M / BM), dim3(256), 0, stream>>>(
        x, w_qkv, b_qkv, qkv, M, 3 * DIM, DIM);

    // Cantor attention: one wave per (b,h,s) -> B*H*S/8 blocks of 256
    attn_cantor<<<dim3((BB * NH * SS) / 8), dim3(256), 0, stream>>>(
        qkv, routes, attn);

    // Output projection: [4096,1024] @ [1024,1024] + b
    gemm_wmma_f32<<<dim3(DIM / BN, M / BM), dim3(256), 0, stream>>>(
        attn, w_out, b_out, (float*)d_out, M, DIM, DIM);
}